// TwoTwoNet_53051436040339
// MI455X (gfx1250) — compile-verified
//
#include <hip/hip_runtime.h>
#include <stdint.h>

// CDNA5 / gfx1250, wave32 only.
typedef float        v2f   __attribute__((ext_vector_type(2)));
typedef float        v8f   __attribute__((ext_vector_type(8)));
typedef unsigned int u32x4 __attribute__((ext_vector_type(4)));
typedef int          i32x8 __attribute__((ext_vector_type(8)));
typedef int          i32x4 __attribute__((ext_vector_type(4)));

#define T_STEPS 512
#define N_DIM   1024
#define CHUNK   (N_DIM / 32)   // per-lane slice of the n dimension

// Full 32-lane f32 sum using V_WMMA_F32_16X16X4_F32.
// A layout (ISA 7.12.2): lanes 0-15 VGPR0 = A[m,0], VGPR1 = A[m,1];
// lanes 16-31 VGPR0 = A[m,2], VGPR1 = A[m,3].  With A = {v, 0} and B = ones,
// D[m,n] = v[m] + v[m+16].  D layout: VGPR r holds D[r,*] (lanes 0-15) and
// D[r+8,*] (lanes 16-31), so summing the 8 D VGPRs per lane gives the low/high
// half-sums; one shfl_xor(16) + add completes the reduction (broadcast to all lanes).
__device__ __forceinline__ float wave_sum32(float v) {
  v2f a; a[0] = v;    a[1] = 0.0f;
  v2f b; b[0] = 1.0f; b[1] = 1.0f;     // all-ones B: layout-independent
  v8f c = {};
  v8f d = __builtin_amdgcn_wmma_f32_16x16x4_f32(false, a, false, b,
                                                (short)0, c, false, false);
  float s = ((d[0] + d[1]) + (d[2] + d[3])) + ((d[4] + d[5]) + (d[6] + d[7]));
  s += __shfl_xor(s, 16, 32);
  return s;
}

// One wave32 per batch element.  Rank-2 recurrence: state is parameterized by
// (alpha, beta); each step evaluates three functionals
//   f_v = sum_i  w_v[i] * clip(d[i]*alpha + e[i]*beta, -1, 1)
// with all per-index data held in VGPRs (5 arrays x 32 elems/lane).
__global__ __launch_bounds__(32) void twotwo_scan(
    const float* __restrict__ x,        const float* __restrict__ state0,
    const float* __restrict__ mask,     const float* __restrict__ w,
    const float* __restrict__ dec_orth, const float* __restrict__ enc_orth,
    const float* __restrict__ decoder,  const float* __restrict__ encoder,
    float* __restrict__ out)
{
  __shared__ float xbuf[T_STEPS];
  __shared__ float obuf[T_STEPS];
  const int b    = blockIdx.x;
  const int lane = threadIdx.x;

  // ---- TDM: async tensor load of x[b, 0:T] (2 KB) into LDS --------------
  // D# per CDNA5 ISA:
  // group0: count=1, lds_addr, global_addr[56:0], type=2
  // group1: data_size=2 (4B), tensor_dim0=512, tensor_dim1=1,
  //         tile_dim0=512, tile_dim1=1, tensor_dim0_stride=512, mask=0
  {
    uint64_t ga = (uint64_t)(uintptr_t)(x + (size_t)b * T_STEPS);
    uint32_t la = (uint32_t)(uintptr_t)(&xbuf[0]);   // AS3 offset in low 32 bits
    u32x4 g0;
    g0[0] = 1u;                                        // count = 1 (valid)
    g0[1] = la;                                        // lds_addr
    g0[2] = (uint32_t)ga;                              // global_addr[31:0]
    g0[3] = ((uint32_t)(ga >> 32) & 0x01FFFFFFu)       // global_addr[56:32]
            | (2u << 30);                              // type = 2 ("image")
    i32x8 g1;
    g1[0] = 0x00020000;                                // data_size=4B, wg_mask=0
    g1[1] = (int)((uint32_t)T_STEPS << 16);            // tensor_dim0[15:0] @63:48
    g1[2] = (int)(1u << 16);                           // tensor_dim0 hi=0, tensor_dim1=1
    g1[3] = (int)((uint32_t)T_STEPS << 16);            // tile_dim0 = 512 @127:112
    g1[4] = 1;                                         // tile_dim1=1, tile_dim2=0
    g1[5] = T_STEPS;                                   // tensor_dim0_stride lo
    g1[6] = 0;                                         // stride hi / dim1_stride lo
    g1[7] = 0;
    i32x4 gz4 = {0, 0, 0, 0};
    i32x8 gz8 = {0, 0, 0, 0, 0, 0, 0, 0};
    // 6-arg form (clang-23 / therock headers): groups 2,3 disabled, extra
    // zero-filled group, cpol = 0.
    __builtin_amdgcn_tensor_load_to_lds(g0, g1, gz4, gz4, gz8, 0);
  }

  // ---- Load per-lane parameter slices into VGPRs; fold init reductions --
  const int base = lane * CHUNK;
  float dA[CHUNK], eA[CHUNK], wdA[CHUNK], weA[CHUNK], woA[CHUNK];
  float pkd = 0.f, pke = 0.f, p0d = 0.f, p0e = 0.f;
  #pragma unroll
  for (int i = 0; i < CHUNK; ++i) {
    const float di = dec_orth[base + i];
    const float ei = enc_orth[base + i];
    const float mi = mask[base + i];
    const float ni = encoder[base + i];
    const float oi = decoder[base + i];
    const float si = state0[base + i];
    dA[i]  = di;       eA[i]  = ei;
    wdA[i] = mi * di;  weA[i] = mi * ei;  woA[i] = mi * oi;
    const float men = mi * ni;
    pkd = fmaf(men, di, pkd);    // (mask*encoder)·d
    pke = fmaf(men, ei, pke);    // (mask*encoder)·e
    p0d = fmaf(si,  di, p0d);    // state0·d
    p0e = fmaf(si,  ei, p0e);    // state0·e
  }
  const float Kd = wave_sum32(pkd);
  const float Ke = wave_sum32(pke);
  float fd = wave_sum32(p0d);    // running  s·d
  float fe = wave_sum32(p0e);    // running  s·e

  const float w00 = w[0], w01 = w[1], w10 = w[2], w11 = w[3];

  __builtin_amdgcn_s_wait_tensorcnt(0);   // xbuf ready

  // ---- Sequential scan over T steps -------------------------------------
  #pragma unroll 1
  for (int t = 0; t < T_STEPS; ++t) {
    const float xt    = xbuf[t];
    const float a     = fmaf(xt, Kd, fd);
    const float cc    = fmaf(xt, Ke, fe);
    const float alpha = fmaf(w00, a, w01 * cc);
    const float beta  = fmaf(w10, a, w11 * cc);
    float ad = 0.f, ae = 0.f, ao = 0.f;
    #pragma unroll
    for (int i = 0; i < CHUNK; ++i) {
      float p = fmaf(dA[i], alpha, eA[i] * beta);
      p = fminf(1.0f, fmaxf(-1.0f, p));        // clip -> v_med3 / max+min
      ad = fmaf(wdA[i], p, ad);
      ae = fmaf(weA[i], p, ae);
      ao = fmaf(woA[i], p, ao);
    }
    fd = wave_sum32(ad);          // s_new·d
    fe = wave_sum32(ae);          // s_new·e
    const float o = wave_sum32(ao);
    if (lane == 0) obuf[t] = o;   // out[b,t]
  }

  __syncthreads();  // single-wave WG: forces dscnt wait before readback

  // ---- Coalesced writeback of the output row ----------------------------
  float* orow = out + (size_t)b * T_STEPS;
  #pragma unroll
  for (int k = 0; k < T_STEPS / 32; ++k)
    orow[lane + 32 * k] = obuf[lane + 32 * k];
}

extern "C" void kernel_launch(void* const* d_in, const int* in_sizes, int n_in,
                              void* d_out, int out_size, void* d_ws, size_t ws_size,
                              hipStream_t stream) {
  (void)n_in; (void)d_ws; (void)ws_size; (void)out_size;
  const float* x        = (const float*)d_in[0];
  const float* state0   = (const float*)d_in[1];
  const float* mask     = (const float*)d_in[2];
  const float* w        = (const float*)d_in[3];
  const float* dec_orth = (const float*)d_in[4];
  const float* enc_orth = (const float*)d_in[5];
  const float* decoder  = (const float*)d_in[6];
  const float* encoder  = (const float*)d_in[7];
  float* out = (float*)d_out;

  const int B = in_sizes[0] / T_STEPS;   // 256
  twotwo_scan<<<B, 32, 0, stream>>>(x, state0, mask, w, dec_orth, enc_orth,
                                    decoder, encoder, out);
}